// GraphCastDecoder_40321152975371
// MI455X (gfx1250) — compile-verified
//
#include <hip/hip_runtime.h>
#include <math.h>
#include <utility>

typedef __attribute__((ext_vector_type(2))) float v2f;
typedef __attribute__((ext_vector_type(8))) float v8f;
typedef __attribute__((ext_vector_type(4))) int v4i;

#if __has_builtin(__builtin_amdgcn_global_load_async_to_lds_b128) && \
    __has_builtin(__builtin_amdgcn_s_wait_asynccnt)
#define USE_ASYNC_LDS 1
#else
#define USE_ASYNC_LDS 0
#endif

namespace {

constexpr int Hd = 128;     // hidden dim
constexpr int AST = 132;    // A-tile LDS row stride in floats (conflict-free b64 A-frags)
constexpr int PSTD = 288;   // W-tile LDS k-pair row stride in dwords (+32-bank shift per pair)
constexpr int ROWS = 128;   // rows (edges / nodes) per workgroup
constexpr int THREADS = 256;

constexpr int LDS_A = 0;
constexpr int LDS_W = ROWS * AST;               // 16896 floats
constexpr int LDS_H = LDS_W + 64 * PSTD;        // + 18432 floats
constexpr int LDS_FLOATS = LDS_H + ROWS * AST;  // 52224 floats = 204 KiB

#if USE_ASYNC_LDS
typedef __attribute__((address_space(1))) v4i* gptr_t;
typedef __attribute__((address_space(3))) v4i* lptr_t;

template <int... J>
__device__ __forceinline__ void async_copy16(gptr_t g, lptr_t l,
                                             std::integer_sequence<int, J...>) {
  (__builtin_amdgcn_global_load_async_to_lds_b128(g, l, J * 16, 0), ...);
}
#endif

// copy 64 consecutive floats global -> LDS (one thread's share of a row)
__device__ __forceinline__ void copy_row64(float* ldsDst, const float* gSrc) {
#if USE_ASYNC_LDS
  gptr_t g = (gptr_t)(void*)gSrc;
  lptr_t l = (lptr_t)(void*)ldsDst;
  async_copy16(g, l, std::make_integer_sequence<int, 16>{});
#else
  const float4* g = (const float4*)gSrc;
  float4* l = (float4*)ldsDst;
#pragma unroll
  for (int j = 0; j < 16; ++j) l[j] = g[j];
#endif
}

__device__ __forceinline__ void stage_fence() {
#if USE_ASYNC_LDS
  __builtin_amdgcn_s_wait_asynccnt(0);
#endif
}

// ---- A-tile staging: 256 threads, 2 threads per row (64 floats each) ----
__device__ __forceinline__ void stage_gather(float* lds, const float* __restrict__ src,
                                             const int* __restrict__ idx, int base) {
  const int t = threadIdx.x;
  const int row = t >> 1;
  const int half = (t & 1) * 64;
  copy_row64(lds + row * AST + half, src + (size_t)idx[base + row] * Hd + half);
}

__device__ __forceinline__ void stage_rows(float* lds, const float* __restrict__ src, int row0) {
  const int t = threadIdx.x;
  const int row = t >> 1;
  const int half = (t & 1) * 64;
  copy_row64(lds + row * AST + half, src + (size_t)(row0 + row) * Hd + half);
}

// ---- W staging, k-pair interleaved: rows k and k+1 of W share LDS row k>>1 (PSTD stride), column n's pair at float 2n ----
__device__ __forceinline__ void stage_w(float* lds, const float* __restrict__ w) {
  const int t = threadIdx.x;
  const int p = t >> 2;            // k-pair row 0..63
  const int q = (t & 3) * 32;      // n base
  const float4* r0 = (const float4*)(w + (size_t)(2 * p) * Hd + q);
  const float4* r1 = (const float4*)(w + (size_t)(2 * p + 1) * Hd + q);
  float2* l = (float2*)(lds + p * PSTD + q * 2);
#pragma unroll
  for (int j = 0; j < 8; ++j) {
    float4 a = r0[j], b = r1[j];
    l[4 * j + 0] = make_float2(a.x, b.x);
    l[4 * j + 1] = make_float2(a.y, b.y);
    l[4 * j + 2] = make_float2(a.z, b.z);
    l[4 * j + 3] = make_float2(a.w, b.w);
  }
}

// ---- one K=128 panel: 16-row strip x 128 cols, f32 WMMA 16x16x4 ----
// A 16x4 frag: lane L<16 holds A[M=L][k0,k0+1]; lanes 16-31 hold k0+2,k0+3  -> one b64
// B 4x16 frag: lane L<16 holds B[k0][N],B[k0+1][N]; lanes 16-31 hold k0+2,k0+3 -> one b64
__device__ __forceinline__ void gemm_k128(const float* __restrict__ A, const float* __restrict__ Wp,
                                          int wave, int l16, int half, v8f acc[8]) {
  const float* arow = A + (wave * 16 + l16) * AST + 2 * half;
  const float* wbase = Wp + half * PSTD + l16 * 2;
#pragma unroll 4
  for (int k0 = 0; k0 < 128; k0 += 4) {
    v2f a = *(const v2f*)(arow + k0);
    const float* wk = wbase + (k0 >> 1) * PSTD;
#pragma unroll
    for (int t = 0; t < 8; ++t) {
      v2f b = *(const v2f*)(wk + t * 32);
      acc[t] = __builtin_amdgcn_wmma_f32_16x16x4_f32(false, a, false, b,
                                                     (short)0, acc[t], false, false);
    }
  }
}

__device__ __forceinline__ void write_hidden_silu(float* Hst, const float* __restrict__ b1,
                                                  int wave, int l16, int half, v8f acc[8]) {
#pragma unroll
  for (int t = 0; t < 8; ++t) {
    const float b1v = b1[t * 16 + l16];
#pragma unroll
    for (int r = 0; r < 8; ++r) {
      float x = acc[t][r] + b1v;
      float s = 1.0f / (1.0f + __expf(-x));
      Hst[(wave * 16 + r + 8 * half) * AST + t * 16 + l16] = x * s;
    }
    acc[t] = (v8f)0.0f;
  }
}

// ============================ edge block ============================
__global__ __launch_bounds__(THREADS, 1)
void edge_kernel(const float* __restrict__ edge_feats,
                 const float* __restrict__ grid_feats,
                 const float* __restrict__ mesh_feats,
                 const float* __restrict__ W1, const float* __restrict__ b1,
                 const float* __restrict__ W2, const float* __restrict__ b2,
                 const float* __restrict__ ln_g, const float* __restrict__ ln_b,
                 const int* __restrict__ src_idx, const int* __restrict__ dst_idx,
                 float* __restrict__ agg) {
  extern __shared__ float lds[];
  float* Ast = lds + LDS_A;
  float* Wst = lds + LDS_W;
  float* Hst = lds + LDS_H;

  const int e0 = blockIdx.x * ROWS;
  const int lane = threadIdx.x & 31;
  const int wave = threadIdx.x >> 5;
  const int l16 = lane & 15;
  const int half = lane >> 4;

  v8f acc[8];
#pragma unroll
  for (int t = 0; t < 8; ++t) acc[t] = (v8f)0.0f;

  // K segment 0: gathered mesh (src) rows vs W1[0:128]
  stage_gather(Ast, mesh_feats, src_idx, e0);
  stage_w(Wst, W1);
  stage_fence();
  __syncthreads();
  gemm_k128(Ast, Wst, wave, l16, half, acc);
  __syncthreads();
  // K segment 1: gathered grid (dst) rows vs W1[128:256]
  stage_gather(Ast, grid_feats, dst_idx, e0);
  stage_w(Wst, W1 + 128 * Hd);
  stage_fence();
  __syncthreads();
  gemm_k128(Ast, Wst, wave, l16, half, acc);
  __syncthreads();
  // K segment 2: edge features (kept in Ast for the residual) vs W1[256:384]
  stage_rows(Ast, edge_feats, e0);
  stage_w(Wst, W1 + 256 * Hd);
  stage_fence();
  __syncthreads();
  gemm_k128(Ast, Wst, wave, l16, half, acc);
  __syncthreads();

  write_hidden_silu(Hst, b1, wave, l16, half, acc);
  stage_w(Wst, W2);
  __syncthreads();
  gemm_k128(Hst, Wst, wave, l16, half, acc);

  // epilogue: +b2, LayerNorm over 128 cols, residual edge feat, scatter-add
  float b2v[8], gv[8], bv[8];
#pragma unroll
  for (int t = 0; t < 8; ++t) {
    const int n = t * 16 + l16;
    b2v[t] = b2[n]; gv[t] = ln_g[n]; bv[t] = ln_b[n];
  }
#pragma unroll
  for (int r = 0; r < 8; ++r) {
    float s = 0.0f, s2 = 0.0f;
#pragma unroll
    for (int t = 0; t < 8; ++t) {
      float v = acc[t][r] + b2v[t];
      acc[t][r] = v;
      s += v; s2 += v * v;
    }
#pragma unroll
    for (int m = 1; m < 16; m <<= 1) {  // reduce within each 16-lane half (one row each)
      s  += __shfl_xor(s,  m, 32);
      s2 += __shfl_xor(s2, m, 32);
    }
    const float mean = s * (1.0f / 128.0f);
    const float var = s2 * (1.0f / 128.0f) - mean * mean;
    const float inv = rsqrtf(var + 1e-5f);
    const int rowLocal = wave * 16 + r + 8 * half;
    const int d = dst_idx[e0 + rowLocal];
    float* aggRow = agg + (size_t)d * Hd;
    const float* eRow = Ast + rowLocal * AST;
#pragma unroll
    for (int t = 0; t < 8; ++t) {
      const int n = t * 16 + l16;
      float v = gv[t] * (acc[t][r] - mean) * inv + bv[t];
      unsafeAtomicAdd(aggRow + n, eRow[n] + v);
    }
  }
}

// ============================ node block ============================
__global__ __launch_bounds__(THREADS, 1)
void node_kernel(const float* __restrict__ grid_feats,
                 const float* __restrict__ agg,
                 const float* __restrict__ W1, const float* __restrict__ b1,
                 const float* __restrict__ W2, const float* __restrict__ b2,
                 const float* __restrict__ ln_g, const float* __restrict__ ln_b,
                 float* __restrict__ out) {
  extern __shared__ float lds[];
  float* Ast = lds + LDS_A;
  float* Wst = lds + LDS_W;
  float* Hst = lds + LDS_H;

  const int n0 = blockIdx.x * ROWS;
  const int lane = threadIdx.x & 31;
  const int wave = threadIdx.x >> 5;
  const int l16 = lane & 15;
  const int half = lane >> 4;

  v8f acc[8];
#pragma unroll
  for (int t = 0; t < 8; ++t) acc[t] = (v8f)0.0f;

  // K segment 1 first: agg rows vs W1[128:256]
  stage_rows(Ast, agg, n0);
  stage_w(Wst, W1 + 128 * Hd);
  stage_fence();
  __syncthreads();
  gemm_k128(Ast, Wst, wave, l16, half, acc);
  __syncthreads();
  // K segment 0 last: grid rows (kept in Ast for residual) vs W1[0:128]
  stage_rows(Ast, grid_feats, n0);
  stage_w(Wst, W1);
  stage_fence();
  __syncthreads();
  gemm_k128(Ast, Wst, wave, l16, half, acc);
  __syncthreads();

  write_hidden_silu(Hst, b1, wave, l16, half, acc);
  stage_w(Wst, W2);
  __syncthreads();
  gemm_k128(Hst, Wst, wave, l16, half, acc);

  float b2v[8], gv[8], bv[8];
#pragma unroll
  for (int t = 0; t < 8; ++t) {
    const int n = t * 16 + l16;
    b2v[t] = b2[n]; gv[t] = ln_g[n]; bv[t] = ln_b[n];
  }
#pragma unroll
  for (int r = 0; r < 8; ++r) {
    float s = 0.0f, s2 = 0.0f;
#pragma unroll
    for (int t = 0; t < 8; ++t) {
      float v = acc[t][r] + b2v[t];
      acc[t][r] = v;
      s += v; s2 += v * v;
    }
#pragma unroll
    for (int m = 1; m < 16; m <<= 1) {
      s  += __shfl_xor(s,  m, 32);
      s2 += __shfl_xor(s2, m, 32);
    }
    const float mean = s * (1.0f / 128.0f);
    const float var = s2 * (1.0f / 128.0f) - mean * mean;
    const float inv = rsqrtf(var + 1e-5f);
    const int rowLocal = wave * 16 + r + 8 * half;
    float* outRow = out + (size_t)(n0 + rowLocal) * Hd;
    const float* gRow = Ast + rowLocal * AST;
#pragma unroll
    for (int t = 0; t < 8; ++t) {
      const int n = t * 16 + l16;
      float v = gv[t] * (acc[t][r] - mean) * inv + bv[t];
      outRow[n] = gRow[n] + v;
    }
  }
}

}  // namespace

extern "C" void kernel_launch(void* const* d_in, const int* in_sizes, int n_in,
                              void* d_out, int out_size, void* d_ws, size_t ws_size,
                              hipStream_t stream) {
  const float* edge_feats = (const float*)d_in[0];
  const float* grid_feats = (const float*)d_in[1];
  const float* mesh_feats = (const float*)d_in[2];
  const float* eW1 = (const float*)d_in[3];
  const float* eb1 = (const float*)d_in[4];
  const float* eW2 = (const float*)d_in[5];
  const float* eb2 = (const float*)d_in[6];
  const float* elg = (const float*)d_in[7];
  const float* elb = (const float*)d_in[8];
  const float* nW1 = (const float*)d_in[9];
  const float* nb1 = (const float*)d_in[10];
  const float* nW2 = (const float*)d_in[11];
  const float* nb2 = (const float*)d_in[12];
  const float* nlg = (const float*)d_in[13];
  const float* nlb = (const float*)d_in[14];
  const int* src_idx = (const int*)d_in[15];
  const int* dst_idx = (const int*)d_in[16];
  float* out = (float*)d_out;
  float* agg = (float*)d_ws;  // [N_GRID, H] f32 scatter target

  const int NE = in_sizes[15];      // 524288 edges
  const int NG = in_sizes[1] / Hd;  // 262144 grid nodes

  (void)hipMemsetAsync(agg, 0, (size_t)NG * Hd * sizeof(float), stream);

  const size_t shmem = (size_t)LDS_FLOATS * sizeof(float);
  edge_kernel<<<NE / ROWS, THREADS, shmem, stream>>>(
      edge_feats, grid_feats, mesh_feats, eW1, eb1, eW2, eb2, elg, elb,
      src_idx, dst_idx, agg);
  node_kernel<<<NG / ROWS, THREADS, shmem, stream>>>(
      grid_feats, agg, nW1, nb1, nW2, nb2, nlg, nlb, out);
}